// HeatMap_74620761801419
// MI455X (gfx1250) — compile-verified
//
#include <hip/hip_runtime.h>
#include <stdint.h>

// Heatmap scatter-max renderer for MI455X (gfx1250).
// Strategy: build each 64-row x 256-col f32 tile in LDS (zero + ds_max_u32
// scatter of positive-float bit patterns), then DMA the finished tile to HBM
// with the CDNA5 Tensor Data Mover (tensor_store_from_lds + s_wait_tensorcnt).
// Output store traffic (256 MB) is the roofline; TDM writes each byte once.

#define IMG        256
#define PATCH      15
#define HALF       7
#define NLMK       30
#define CHUNK_ROWS 64
#define CHUNKS     (IMG / CHUNK_ROWS)          // 4 chunks per image
#define PAIRS      (NLMK * PATCH * PATCH)      // 6750 (landmark, pixel) pairs
#define TPB        256

typedef unsigned int  uint32x4 __attribute__((ext_vector_type(4)));
typedef int           int32x4  __attribute__((ext_vector_type(4)));
typedef int           int32x8  __attribute__((ext_vector_type(8)));

__global__ __launch_bounds__(TPB) void heatmap_tdm_kernel(
    const float* __restrict__ lmk,   // [N, NLMK, 2]
    float* __restrict__ out)         // [N, IMG, IMG]
{
    __shared__ unsigned int tile[CHUNK_ROWS * IMG];   // 64 KB, f32 bit patterns

    const int tid  = threadIdx.x;
    const int blk  = blockIdx.x;
    const int n    = blk >> 2;            // image index
    const int c    = blk & (CHUNKS - 1);  // chunk index
    const int row0 = c * CHUNK_ROWS;

    // ---- 1) zero the LDS tile (ds_store_b128, 16 iters/thread) ----
    {
        uint32x4* p = (uint32x4*)tile;
        for (int i = tid; i < (CHUNK_ROWS * IMG) / 4; i += TPB)
            p[i] = (uint32x4)0;
    }
    __syncthreads();

    // ---- 2) scatter-max patch values into LDS ----
    const float lo = (float)HALF;                 // 7
    const float hi = (float)(IMG - 1 - HALF);     // 248
    for (int k = tid; k < PAIRS; k += TPB) {
        const int l  = k / (PATCH * PATCH);
        const int p  = k - l * (PATCH * PATCH);
        const int q  = p / PATCH;
        const int o0 = q - HALF;                  // row offset, -7..7
        const int o1 = (p - q * PATCH) - HALF;    // col offset, -7..7

        float a = lmk[(n * NLMK + l) * 2 + 0];
        float b = lmk[(n * NLMK + l) * 2 + 1];
        a = fminf(fmaxf(a, lo), hi);
        b = fminf(fmaxf(b, lo), hi);
        if (a == lo && b == lo) continue;         // reference skips this landmark

        const float fa = floorf(a), fb = floorf(b);
        const int row = (int)fa + o0;             // always in [0,255]
        if (row < row0 || row >= row0 + CHUNK_ROWS) continue;
        const int col = (int)fb + o1;

        const float dx = (float)o0 - (a - fa);
        const float dy = (float)o1 - (b - fb);
        const float v  = 1.0f / sqrtf(1.0f + dx * dx + dy * dy + 1e-6f);
        // positive floats: IEEE order == unsigned-int order; 0x0 init is identity
        atomicMax(&tile[(row - row0) * IMG + col], __float_as_uint(v));
    }
    __syncthreads();

    // ---- 3) TDM store: LDS tile -> global, one wave issues the DMA ----
    const int wave = __builtin_amdgcn_readfirstlane(tid >> 5);
    if (wave == 0) {
        const uint64_t gaddr = (uint64_t)(uintptr_t)out +
            ((uint64_t)n * IMG * IMG + (uint64_t)row0 * IMG) * 4u;
        const uint32_t lds_addr = (uint32_t)(uintptr_t)&tile[0]; // LDS byte offset

        // D# group 0: count=1 | lds_addr | global_addr[56:0] | type=2
        uint32x4 g0;
        g0.x = 1u;                                        // count=1, user mode
        g0.y = lds_addr;                                  // bits 63:32
        g0.z = (uint32_t)gaddr;                           // global_addr[31:0]
        g0.w = (uint32_t)(gaddr >> 32) | (2u << 30);      // ga[56:32] | type=2

        // D# group 1: data_size=4B, tensor 256x64, tile 256x64, stride0=256
        int32x8 g1;
        g1[0] = 0x00020000;                  // wg_mask=0, data_size=2 (4 bytes)
        g1[1] = (int)((unsigned)IMG << 16);  // tensor_dim0 = 256 (bits 63:48)
        g1[2] = (int)((unsigned)CHUNK_ROWS << 16); // tensor_dim1 = 64 (95:80)
        g1[3] = (int)((unsigned)IMG << 16);  // tile_dim0 = 256 (bits 127:112)
        g1[4] = CHUNK_ROWS;                  // tile_dim1 = 64, tile_dim2 = 0
        g1[5] = IMG;                         // tensor_dim0_stride = 256
        g1[6] = 0;                           // stride0 hi / tensor_dim1_stride
        g1[7] = 0;

        const int32x4 gz = (int32x4)0;       // groups 2/3 unused (2D tensor)
#if defined(__clang_major__) && (__clang_major__ >= 23)
        __builtin_amdgcn_tensor_store_from_lds(g0, g1, gz, gz, (int32x8)0, 0);
#else
        __builtin_amdgcn_tensor_store_from_lds(g0, g1, gz, gz, 0);
#endif
        __builtin_amdgcn_s_wait_tensorcnt(0);  // LDS must stay live until done
    }
}

extern "C" void kernel_launch(void* const* d_in, const int* in_sizes, int n_in,
                              void* d_out, int out_size, void* d_ws, size_t ws_size,
                              hipStream_t stream) {
    const float* lmk = (const float*)d_in[0];   // [N, 30, 2] f32
    // d_in[1] (offsets) is a constant -7..7 product grid; computed analytically.
    float* out = (float*)d_out;                 // [N, 256, 256] f32

    const int N = in_sizes[0] / (NLMK * 2);     // 1024
    heatmap_tdm_kernel<<<dim3(N * CHUNKS), dim3(TPB), 0, stream>>>(lmk, out);
}